// ESN_28518582845836
// MI455X (gfx1250) — compile-verified
//
#include <hip/hip_runtime.h>
#include <math.h>

// ESN forward (collapsed): out[m, n] = 0.5 * tanh( sum_k x[m,k] * w_in[n,k] )
// M = 32*4096 = 131072, N = 512, K = 64, all fp32.
// Store-bandwidth bound (~268 MB output); WMMA F32_16x16x4 for the GEMM core.

typedef __attribute__((ext_vector_type(2))) float v2f;
typedef __attribute__((ext_vector_type(8))) float v8f;

#define ESN_M 131072
#define ESN_N 512
#define ESN_K 64
#define NT_PER_WAVE 8       // 8 N-tiles of 16 -> 128 cols per wave
#define WAVES_PER_BLOCK 8   // 256 threads

__global__ __launch_bounds__(256)
void esn_wmma_kernel(const float* __restrict__ x,
                     const float* __restrict__ w,
                     float* __restrict__ out)
{
    const int lane        = threadIdx.x & 31;
    const int waveInBlock = threadIdx.x >> 5;
    const int waveId      = blockIdx.x * WAVES_PER_BLOCK + waveInBlock;

    const int nBandsPerM  = ESN_N / (16 * NT_PER_WAVE);   // 4
    const int nBand       = waveId % nBandsPerM;
    const int mTile       = waveId / nBandsPerM;

    const int l16  = lane & 15;
    const int half = lane >> 4;

    const int mBase = mTile * 16;
    const int nBase = nBand * (16 * NT_PER_WAVE);

    // ---- Preload A tile (16 x 64 fp32) into registers -----------------------
    // A 16x4 f32 layout: lanes 0-15 row M=l16 K={0,1}, lanes 16-31 same rows
    // K={2,3}; per K-step j the lane needs x[row][4j + 2*half + {0,1}].
    const float* aPtr = x + (size_t)(mBase + l16) * ESN_K + 2 * half;
    v2f a[ESN_K / 4];
#pragma unroll
    for (int j = 0; j < ESN_K / 4; ++j)
        a[j] = *(const v2f*)(aPtr + 4 * j);

    // ---- 8 accumulators: 16x16 each -----------------------------------------
    v8f acc[NT_PER_WAVE];
#pragma unroll
    for (int nt = 0; nt < NT_PER_WAVE; ++nt) {
        v8f c = {};
        // B 4x16 (K x N) with B[k][n] = w_in[n][k]: lane's column n = tile + l16,
        // per K-step j the lane needs w[n][4j + 2*half + {0,1}] -> contiguous f2.
        const float* bPtr = w + (size_t)(nBase + nt * 16 + l16) * ESN_K + 2 * half;
#pragma unroll
        for (int j = 0; j < ESN_K / 4; ++j) {
            v2f b = *(const v2f*)(bPtr + 4 * j);
            // 8 args: (neg_a, A, neg_b, B, c_mod, C, reuse_a, reuse_b)
            c = __builtin_amdgcn_wmma_f32_16x16x4_f32(
                    false, a[j], false, b, (short)0, c, false, false);
        }
        acc[nt] = c;
    }

    // ---- Epilogue: 0.5 * tanh, coalesced fp32 stores ------------------------
    // C/D layout: VGPR v -> row (v + 8*half), col = l16 within tile.
#pragma unroll
    for (int nt = 0; nt < NT_PER_WAVE; ++nt) {
        const int col = nBase + nt * 16 + l16;
#pragma unroll
        for (int v = 0; v < 8; ++v) {
            const int row = mBase + v + 8 * half;
            out[(size_t)row * ESN_N + col] = 0.5f * tanhf(acc[nt][v]);
        }
    }
}

extern "C" void kernel_launch(void* const* d_in, const int* in_sizes, int n_in,
                              void* d_out, int out_size, void* d_ws, size_t ws_size,
                              hipStream_t stream)
{
    (void)in_sizes; (void)n_in; (void)out_size; (void)d_ws; (void)ws_size;
    const float* x = (const float*)d_in[0];   // [131072, 64]
    const float* w = (const float*)d_in[1];   // [512, 64]
    // d_in[2] (diagonal d) is mathematically dead: state is always zero.
    float* out = (float*)d_out;               // [131072, 512]

    const int totalWaves = (ESN_M / 16) * (ESN_N / (16 * NT_PER_WAVE)); // 32768
    const int blocks     = totalWaves / WAVES_PER_BLOCK;                // 4096
    esn_wmma_kernel<<<blocks, WAVES_PER_BLOCK * 32, 0, stream>>>(x, w, out);
}